// PrimitivesGatingFF_60309930770443
// MI455X (gfx1250) — compile-verified
//
#include <hip/hip_runtime.h>
#include <hip/hip_bf16.h>

// ---------------------------------------------------------------------------
// PrimitivesGatingFF for MI455X (gfx1250, wave32, WMMA f32 16x16x4)
//
// B=4, T=4096 -> BT=16384 rows, D=1024, P=64, H=64, K_SEL=8.
// Memory-bound (~210 MB traffic, ~6.6 GFLOP) -> keep fp32, use
// V_WMMA_F32_16X16X4_F32 for all three big GEMMs.
// ---------------------------------------------------------------------------

typedef float v2f __attribute__((ext_vector_type(2)));
typedef float v8f __attribute__((ext_vector_type(8)));

#define BT    16384
#define D     1024
#define P     64
#define H     64
#define KSEL  8

// workspace layout (float offsets)
#define OFF_GATE   ((size_t)0)                    // [16384][64] gate_probs
#define OFF_PART   (OFF_GATE + (size_t)BT * P)    // [128][64] block partial col sums
#define OFF_H      (OFF_PART + (size_t)128 * P)   // [16384][64] relu(z @ Wd^T)
#define OFF_BLND   (OFF_H    + (size_t)BT * H)    // [1024] blended
#define OFF_IDX    (OFF_BLND + (size_t)D)         // [8] top indices (int)

__device__ __forceinline__ v8f wmma4f32(v2f a, v2f b, v8f c) {
    // D = A(16x4) * B(4x16) + C(16x16), all fp32
    return __builtin_amdgcn_wmma_f32_16x16x4_f32(
        /*neg_a=*/false, a, /*neg_b=*/false, b,
        /*c_mod=*/(short)0, c, /*reuse_a=*/false, /*reuse_b=*/false);
}

// ---------------------------------------------------------------------------
// Kernel A: gate logits (WMMA) + softmax + gate_probs + per-block column sums
// grid 128 blocks x 256 threads; each wave: 16 rows x full P=64
// ---------------------------------------------------------------------------
__global__ void __launch_bounds__(256)
k_gate(const float* __restrict__ x, const float* __restrict__ gate_w,
       const float* __restrict__ gate_b, float* __restrict__ ws) {
    __shared__ __align__(16) float lds[128 * P];   // [128 rows][64 p]

    float* gate_probs = ws + OFF_GATE;
    float* partials   = ws + OFF_PART;

    const int lane = threadIdx.x & 31;
    const int wv   = threadIdx.x >> 5;     // 0..7
    const int half = lane >> 4;            // 0/1 -> k offset 0/2
    const int lr   = lane & 15;            // 0..15

    const int rowbase = blockIdx.x * 128 + wv * 16;
    const int m       = rowbase + lr;      // A row handled by this lane
    const float* xrow = x + (size_t)m * D;

    v8f acc[4];
#pragma unroll
    for (int t = 0; t < 4; ++t) acc[t] = (v8f)0.0f;

    for (int k = 0; k < D; k += 4) {
        const int kc = k + 2 * half;
        v2f a = *(const v2f*)(xrow + kc);
#pragma unroll
        for (int t = 0; t < 4; ++t) {
            // B[k][p] = gate_w[p][k]
            v2f b = *(const v2f*)(gate_w + (size_t)(t * 16 + lr) * D + kc);
            acc[t] = wmma4f32(a, b, acc[t]);
        }
    }

    // add bias, dump logits to LDS as [row][p]
    float* myrows = lds + (wv * 16) * P;
#pragma unroll
    for (int t = 0; t < 4; ++t) {
        const float gb = gate_b[t * 16 + lr];
#pragma unroll
        for (int v = 0; v < 8; ++v) {
            myrows[(v + 8 * half) * P + t * 16 + lr] = acc[t][v] + gb;
        }
    }
    __syncthreads();

    // per-row softmax over P=64 (threads 0..127, one row each)
    if (threadIdx.x < 128) {
        float* rp = lds + threadIdx.x * P;
        float mx = rp[0];
        for (int p = 1; p < P; ++p) mx = fmaxf(mx, rp[p]);
        float s = 0.0f;
        for (int p = 0; p < P; ++p) { float e = __expf(rp[p] - mx); rp[p] = e; s += e; }
        const float inv = 1.0f / s;
        for (int p = 0; p < P; ++p) rp[p] *= inv;
    }
    __syncthreads();

    // coalesced writeout of this block's [128][64] probs
    float* gdst = gate_probs + (size_t)blockIdx.x * 128 * P;
#pragma unroll
    for (int i = 0; i < 32; ++i) {
        const int idx = threadIdx.x + i * 256;
        gdst[idx] = lds[idx];
    }

    // per-block column sums (for mean over all rows)
    if (threadIdx.x < P) {
        float s = 0.0f;
        for (int r = 0; r < 128; ++r) s += lds[r * P + threadIdx.x];
        partials[(size_t)blockIdx.x * P + threadIdx.x] = s;
    }
}

// ---------------------------------------------------------------------------
// Kernel B: mean reduce + top-8 + blend softmax column sums + blended[D]
// single block, 256 threads
// ---------------------------------------------------------------------------
__global__ void __launch_bounds__(256)
k_select(const float* __restrict__ primitives, const float* __restrict__ blend,
         float* __restrict__ ws) {
    __shared__ float meanp[P];
    __shared__ float colw[KSEL];
    __shared__ int   sidx[KSEL];

    const float* partials = ws + OFF_PART;
    float*       blended  = ws + OFF_BLND;
    int*         widx     = (int*)(ws + OFF_IDX);

    const int t = threadIdx.x;
    if (t < P) {
        float s = 0.0f;
        for (int b = 0; b < 128; ++b) s += partials[(size_t)b * P + t];
        meanp[t] = s * (1.0f / (float)BT);
    }
    __syncthreads();

    if (t == 0) {
        // top-8 (strict > => lowest index wins ties, matching jax top_k)
        for (int k = 0; k < KSEL; ++k) {
            float best = -3.0e38f; int bi = 0;
            for (int p = 0; p < P; ++p)
                if (meanp[p] > best) { best = meanp[p]; bi = p; }
            sidx[k] = bi;
            widx[k] = bi;
            meanp[bi] = -3.0e38f;
        }
        // blend softmax, column sums: w[j] = sum_i softmax(blend[i,:])[j]
        float cw[KSEL];
#pragma unroll
        for (int j = 0; j < KSEL; ++j) cw[j] = 0.0f;
        for (int i = 0; i < KSEL; ++i) {
            float mx = blend[i * KSEL];
            for (int j = 1; j < KSEL; ++j) mx = fmaxf(mx, blend[i * KSEL + j]);
            float e[KSEL], s = 0.0f;
            for (int j = 0; j < KSEL; ++j) { e[j] = __expf(blend[i * KSEL + j] - mx); s += e[j]; }
            const float inv = 1.0f / s;
            for (int j = 0; j < KSEL; ++j) cw[j] += e[j] * inv;
        }
#pragma unroll
        for (int j = 0; j < KSEL; ++j) colw[j] = cw[j];
    }
    __syncthreads();

    // blended[d] = sum_k colw[k] * primitives[idx[k]][d]
    for (int d = t; d < D; d += 256) {
        float s = 0.0f;
#pragma unroll
        for (int k = 0; k < KSEL; ++k)
            s += colw[k] * primitives[(size_t)sidx[k] * D + d];
        blended[d] = s;
    }
}

// ---------------------------------------------------------------------------
// Kernel C1: z = x + sel_probs@selected + 0.1*blended ; h = relu(z @ Wd^T)
// grid 128 x 256; per wave: 16 rows x H=64 via WMMA, K=1024
// ---------------------------------------------------------------------------
__global__ void __launch_bounds__(256)
k_down(const float* __restrict__ x, const float* __restrict__ primitives,
       const float* __restrict__ wd, float* __restrict__ ws) {
    __shared__ __align__(16) float sprim[KSEL * D];  // 32 KB: selected rows
    __shared__ __align__(16) float sblnd[D];         // 4 KB

    const float* gate_probs = ws + OFF_GATE;
    const float* blended    = ws + OFF_BLND;
    const int*   widx       = (const int*)(ws + OFF_IDX);
    float*       hbuf       = ws + OFF_H;

    int idx8[KSEL];
#pragma unroll
    for (int k = 0; k < KSEL; ++k) idx8[k] = widx[k];

    // stage selected primitives + blended in LDS (uniform loops, no divergence)
#pragma unroll
    for (int i = 0; i < (KSEL * D) / 256; ++i) {
        const int e = threadIdx.x + i * 256;
        sprim[e] = primitives[(size_t)idx8[e >> 10] * D + (e & (D - 1))];
    }
#pragma unroll
    for (int i = 0; i < D / 256; ++i) {
        const int e = threadIdx.x + i * 256;
        sblnd[e] = blended[e];
    }
    __syncthreads();

    const int lane = threadIdx.x & 31;
    const int wv   = threadIdx.x >> 5;
    const int half = lane >> 4;
    const int lr   = lane & 15;

    const int rowbase = blockIdx.x * 128 + wv * 16;
    const int m       = rowbase + lr;

    float selp[KSEL];
#pragma unroll
    for (int k = 0; k < KSEL; ++k)
        selp[k] = gate_probs[(size_t)m * P + idx8[k]];

    const float* xrow = x + (size_t)m * D;

    v8f acc[4];
#pragma unroll
    for (int t = 0; t < 4; ++t) acc[t] = (v8f)0.0f;

    for (int k = 0; k < D; k += 4) {
        const int kc = k + 2 * half;
        v2f xx = *(const v2f*)(xrow + kc);
        v2f bl = *(const v2f*)(sblnd + kc);
        float z0 = xx.x + 0.1f * bl.x;
        float z1 = xx.y + 0.1f * bl.y;
#pragma unroll
        for (int j = 0; j < KSEL; ++j) {
            v2f pp = *(const v2f*)(sprim + j * D + kc);
            z0 = fmaf(selp[j], pp.x, z0);
            z1 = fmaf(selp[j], pp.y, z1);
        }
        v2f a = {z0, z1};
#pragma unroll
        for (int t = 0; t < 4; ++t) {
            // B[k][hcol] = wd[hcol][k]
            v2f b = *(const v2f*)(wd + (size_t)(t * 16 + lr) * D + kc);
            acc[t] = wmma4f32(a, b, acc[t]);
        }
    }

    // relu + store h[16 rows][64]
#pragma unroll
    for (int t = 0; t < 4; ++t) {
#pragma unroll
        for (int v = 0; v < 8; ++v) {
            const int row = rowbase + v + 8 * half;
            hbuf[(size_t)row * H + t * 16 + lr] = fmaxf(acc[t][v], 0.0f);
        }
    }
}

// ---------------------------------------------------------------------------
// Kernel C2: out = h @ Wu^T  (M=16384, K=64, N=1024)
// grid (1024, 2) x 256; per wave: 16 rows x 64 out cols
// ---------------------------------------------------------------------------
__global__ void __launch_bounds__(256)
k_up(const float* __restrict__ wu, const float* __restrict__ ws,
     float* __restrict__ out) {
    const float* hbuf = ws + OFF_H;

    const int lane = threadIdx.x & 31;
    const int wv   = threadIdx.x >> 5;
    const int half = lane >> 4;
    const int lr   = lane & 15;

    const int rowbase = blockIdx.x * 16;          // 1024 row tiles
    const int m       = rowbase + lr;
    const int n0base  = blockIdx.y * 512 + wv * 64;

    const float* hrow = hbuf + (size_t)m * H;

    v8f acc[4];
#pragma unroll
    for (int t = 0; t < 4; ++t) acc[t] = (v8f)0.0f;

    for (int k = 0; k < H; k += 4) {
        const int kc = k + 2 * half;
        v2f a = *(const v2f*)(hrow + kc);
#pragma unroll
        for (int t = 0; t < 4; ++t) {
            // B[k][d] = wu[d][k]
            v2f b = *(const v2f*)(wu + (size_t)(n0base + t * 16 + lr) * H + kc);
            acc[t] = wmma4f32(a, b, acc[t]);
        }
    }

#pragma unroll
    for (int t = 0; t < 4; ++t) {
#pragma unroll
        for (int v = 0; v < 8; ++v) {
            const int row = rowbase + v + 8 * half;
            out[(size_t)row * D + n0base + t * 16 + lr] = acc[t][v];
        }
    }
}

// ---------------------------------------------------------------------------
extern "C" void kernel_launch(void* const* d_in, const int* in_sizes, int n_in,
                              void* d_out, int out_size, void* d_ws, size_t ws_size,
                              hipStream_t stream) {
    const float* x          = (const float*)d_in[0];  // [4,4096,1024]
    const float* primitives = (const float*)d_in[1];  // [64,1024]
    const float* gate_w     = (const float*)d_in[2];  // [64,1024]
    const float* gate_b     = (const float*)d_in[3];  // [64]
    const float* blend      = (const float*)d_in[4];  // [8,8]
    const float* wd         = (const float*)d_in[5];  // [64,1024]
    const float* wu         = (const float*)d_in[6];  // [1024,64]
    float* out = (float*)d_out;
    float* ws  = (float*)d_ws;

    k_gate  <<<dim3(128), dim3(256), 0, stream>>>(x, gate_w, gate_b, ws);
    k_select<<<dim3(1),   dim3(256), 0, stream>>>(primitives, blend, ws);
    k_down  <<<dim3(128), dim3(256), 0, stream>>>(x, primitives, wd, ws);
    k_up    <<<dim3(1024, 2), dim3(256), 0, stream>>>(wu, ws, out);
}